// RARL2OptimizerAD_52020643889816
// MI455X (gfx1250) — compile-verified
//
#include <hip/hip_runtime.h>
#include <math.h>
#include <stdint.h>

// Problem sizes (fixed by the reference)
#define Nn 512
#define Pp 128
#define Mm 128
#define NSTEIN 50
#define NFREQ 50
#define NSLOT 25

static const int NN = Nn * Nn;       // 262144
static const int PN = Pp * Nn;       // 65536
static const int NM = Nn * Mm;       // 65536
static const int PM = Pp * Mm;       // 16384
static const int NAUG = Nn + Mm;     // 640
static const size_t SLOT_F = (size_t)2 * Nn * NAUG + (size_t)2 * Pp * Mm;

typedef float v2f __attribute__((ext_vector_type(2)));
typedef float v8f __attribute__((ext_vector_type(8)));

// ---------------------------------------------------------------------------
// CDNA5 async global->LDS copy (ASYNCcnt-tracked), 128b per lane.
// LDS address = low 32 bits of the generic shared-memory pointer
// (ISA: LDS aperture maps addr[31:0] directly to the LDS byte address).
// ---------------------------------------------------------------------------
__device__ __forceinline__ void async_b128(uint32_t ldsAddr, const float* g) {
  asm volatile("global_load_async_to_lds_b128 %0, %1, off"
               :: "v"(ldsAddr), "v"((unsigned long long)(uintptr_t)g)
               : "memory");
}
__device__ __forceinline__ void wait_async_le4() {
  asm volatile("s_wait_asynccnt 0x4" ::: "memory");
}
__device__ __forceinline__ void wait_async_0() {
  asm volatile("s_wait_asynccnt 0x0" ::: "memory");
}

// ---------------------------------------------------------------------------
// Utility kernels
// ---------------------------------------------------------------------------
__global__ void k_d2f(const double* __restrict__ src, float* __restrict__ dst, int n) {
  int i = blockIdx.x * 256 + threadIdx.x;
  if (i < n) dst[i] = (float)src[i];
}

__global__ void k_buildV(const double* __restrict__ params, float* __restrict__ vRe,
                         float* __restrict__ vIm, int np) {
  int i = blockIdx.x * 256 + threadIdx.x;
  if (i < np) {
    vRe[i] = (float)params[i];
    vIm[i] = (float)params[np + i];
  }
}

__global__ void k_zero(float* __restrict__ p, int n) {
  int i = blockIdx.x * 256 + threadIdx.x;
  if (i < n) p[i] = 0.0f;
}

__global__ void k_addeye(float* __restrict__ gRe) {
  int i = blockIdx.x * 256 + threadIdx.x;
  if (i < Nn) gRe[i * Nn + i] += 1.0f;
}

// ---------------------------------------------------------------------------
// Complex GEMM via V_WMMA_F32_16X16X4_F32, fed by async global->LDS copies.
// Output[m][n] = outSign * ( sum_k opA(A)[m][k] * B[k][n] + (add? add[m][n]:0) )
//   CONJT=false : opA(A)[m][k] = A[m*lda+k]           (A is MxK row-major)
//   CONJT=true  : opA(A)[m][k] = conj(A[k*lda+m])     (A stored KxM row-major)
// WG = 256 threads = 8 waves in a 4x2 (MxN) arrangement of 16x16 tiles,
// WG tile 64x32, K-panel 16, double-buffered in LDS (24KB).
// grid.x = N/32, grid.y = M/64; K must be a multiple of 16 (true here).
// ---------------------------------------------------------------------------
template <bool CONJT>
__global__ void __launch_bounds__(256)
k_cgemm(const float* __restrict__ aRe, const float* __restrict__ aIm, int lda,
        const float* __restrict__ bRe, const float* __restrict__ bIm, int ldb,
        float* __restrict__ oRe, float* __restrict__ oIm, int ldo,
        const float* __restrict__ addRe, const float* __restrict__ addIm,
        int K, float outSign) {
  // Per double-buffer half: A-re 1024 | A-im 1024 | B-re 512 | B-im 512 floats
  __shared__ float smem[2 * 3072];

  const int tid = threadIdx.x;
  const int lane = tid & 31;
  const int wave = tid >> 5;
  const int m16 = lane & 15;
  const int half = lane >> 4;
  const int row0wg = blockIdx.y * 64;
  const int col0wg = blockIdx.x * 32;
  const int wrow = (wave & 3) * 16;   // wave row offset within WG tile
  const int wcol = (wave >> 2) * 16;  // wave col offset within WG tile
  const uint32_t ldsBase = (uint32_t)(uintptr_t)(&smem[0]);

  // Per-thread b128 chunk assignments (each chunk = 4 consecutive floats).
  const int aM = tid >> 2, aK4 = (tid & 3) * 4;      // A panel, [m][k] layout
  const int atK = tid >> 4, atM4 = (tid & 15) * 4;   // A^T panel, [k][m] layout
  const int bChunk = wave * 16 + m16;                // 0..127, lanes 0-15 only
  const int bK = bChunk >> 3, bN4 = (bChunk & 7) * 4;

  v8f accRe = {};
  v8f accIm = {};
  if (addRe != nullptr) {
#pragma unroll
    for (int r = 0; r < 8; ++r) {
      int idx = (row0wg + wrow + r + 8 * half) * ldo + col0wg + wcol + m16;
      accRe[r] = addRe[idx];
      accIm[r] = addIm[idx];
    }
  }

  const int nPanels = K / 16;

  // Issue one K-panel's async copies into buffer `buf` (4 per wave, uniform).
  auto issue = [&](int buf, int k0) {
    const uint32_t base = ldsBase + (uint32_t)buf * 3072u * 4u;
    if (CONJT) {
      // store K-major: panel[k][m] <- A[k0+k][row0wg+m] (contiguous in m)
      async_b128(base + (uint32_t)(atK * 64 + atM4) * 4u,
                 aRe + (size_t)(k0 + atK) * lda + row0wg + atM4);
      async_b128(base + 4096u + (uint32_t)(atK * 64 + atM4) * 4u,
                 aIm + (size_t)(k0 + atK) * lda + row0wg + atM4);
    } else {
      // store M-major: panel[m][k] <- A[row0wg+m][k0+k] (contiguous in k)
      async_b128(base + (uint32_t)(aM * 16 + aK4) * 4u,
                 aRe + (size_t)(row0wg + aM) * lda + k0 + aK4);
      async_b128(base + 4096u + (uint32_t)(aM * 16 + aK4) * 4u,
                 aIm + (size_t)(row0wg + aM) * lda + k0 + aK4);
    }
    if (lane < 16) {  // half-EXEC: still one instruction per wave per plane
      async_b128(base + 8192u + (uint32_t)(bK * 32 + bN4) * 4u,
                 bRe + (size_t)(k0 + bK) * ldb + col0wg + bN4);
      async_b128(base + 10240u + (uint32_t)(bK * 32 + bN4) * 4u,
                 bIm + (size_t)(k0 + bK) * ldb + col0wg + bN4);
    }
  };

  issue(0, 0);
  for (int p = 0; p < nPanels; ++p) {
    if (p + 1 < nPanels) {
      issue((p + 1) & 1, (p + 1) * 16);
      wait_async_le4();  // panel p (this wave's oldest 4 copies) complete
    } else {
      wait_async_0();
    }
    __syncthreads();  // all waves' copies for panel p have landed

    const float* lA = smem + (p & 1) * 3072;
    const float* lAi = lA + 1024;
    const float* lB = lA + 2048;
    const float* lBi = lA + 2560;

#pragma unroll
    for (int kk = 0; kk < 16; kk += 4) {
      const int kl = kk + 2 * half;  // per-lane K pair base (16x4 f32 layout)
      v2f ar, aiE;
      if (CONJT) {
        ar[0] = lA[kl * 64 + wrow + m16];
        ar[1] = lA[(kl + 1) * 64 + wrow + m16];
        aiE[0] = -lAi[kl * 64 + wrow + m16];
        aiE[1] = -lAi[(kl + 1) * 64 + wrow + m16];
      } else {
        const float* pr = lA + (wrow + m16) * 16 + kl;
        const float* pi = lAi + (wrow + m16) * 16 + kl;
        ar[0] = pr[0]; ar[1] = pr[1];
        aiE[0] = pi[0]; aiE[1] = pi[1];
      }
      v2f aiN; aiN[0] = -aiE[0]; aiN[1] = -aiE[1];

      v2f br, bi;
      br[0] = lB[kl * 32 + wcol + m16];
      br[1] = lB[(kl + 1) * 32 + wcol + m16];
      bi[0] = lBi[kl * 32 + wcol + m16];
      bi[1] = lBi[(kl + 1) * 32 + wcol + m16];

      // Complex MAC: re += Ar*Br - AiE*Bi ; im += Ar*Bi + AiE*Br
      accRe = __builtin_amdgcn_wmma_f32_16x16x4_f32(false, ar,  false, br, (short)0, accRe, false, false);
      accRe = __builtin_amdgcn_wmma_f32_16x16x4_f32(false, aiN, false, bi, (short)0, accRe, false, false);
      accIm = __builtin_amdgcn_wmma_f32_16x16x4_f32(false, ar,  false, bi, (short)0, accIm, false, false);
      accIm = __builtin_amdgcn_wmma_f32_16x16x4_f32(false, aiE, false, br, (short)0, accIm, false, false);
    }
    __syncthreads();  // buffer p&1 free before it is refilled next iteration
  }

#pragma unroll
  for (int r = 0; r < 8; ++r) {
    int idx = (row0wg + wrow + r + 8 * half) * ldo + col0wg + wcol + m16;
    oRe[idx] = outSign * accRe[r];
    oIm[idx] = outSign * accIm[r];
  }
}

// ---------------------------------------------------------------------------
// In-place complex Cholesky (upper) of 512x512 Hermitian positive definite G.
// ---------------------------------------------------------------------------
__global__ void k_chol(float* __restrict__ gRe, float* __restrict__ gIm) {
  __shared__ float rowRe[Nn];
  __shared__ float rowIm[Nn];
  __shared__ float dinv;
  const int tid = threadIdx.x;
  const int nt = blockDim.x;
  for (int k = 0; k < Nn; ++k) {
    if (tid == 0) {
      float d = sqrtf(fmaxf(gRe[k * Nn + k], 1e-30f));
      gRe[k * Nn + k] = d;
      gIm[k * Nn + k] = 0.0f;
      dinv = 1.0f / d;
    }
    __syncthreads();
    for (int j = k + 1 + tid; j < Nn; j += nt) {
      gRe[k * Nn + j] *= dinv;
      gIm[k * Nn + j] *= dinv;
    }
    __syncthreads();
    for (int j = k + tid; j < Nn; j += nt) {
      rowRe[j] = gRe[k * Nn + j];
      rowIm[j] = gIm[k * Nn + j];
    }
    __syncthreads();
    const int r = Nn - 1 - k;
    for (int t = tid; t < r * r; t += nt) {
      int i = k + 1 + t / r;
      int j = k + 1 + t % r;
      if (j >= i) {
        float ar = rowRe[i], ai = -rowIm[i];
        float br = rowRe[j], bi = rowIm[j];
        gRe[i * Nn + j] -= ar * br - ai * bi;
        gIm[i * Nn + j] -= ar * bi + ai * br;
      }
    }
    __syncthreads();
  }
}

// ---------------------------------------------------------------------------
// X = R^{-1} (upper-triangular), one thread per column.
// ---------------------------------------------------------------------------
__global__ void k_trinv(const float* __restrict__ rRe, const float* __restrict__ rIm,
                        float* __restrict__ xRe, float* __restrict__ xIm) {
  int j = blockIdx.x * blockDim.x + threadIdx.x;
  if (j >= Nn) return;
  for (int i = Nn - 1; i >= 0; --i) {
    if (i > j) { xRe[i * Nn + j] = 0.0f; xIm[i * Nn + j] = 0.0f; continue; }
    float sr = (i == j) ? 1.0f : 0.0f, si = 0.0f;
    for (int k = i + 1; k <= j; ++k) {
      float ar = rRe[i * Nn + k], ai = rIm[i * Nn + k];
      float br = xRe[k * Nn + j], bi = xIm[k * Nn + j];
      sr -= ar * br - ai * bi;
      si -= ar * bi + ai * br;
    }
    float d = rRe[i * Nn + i];
    xRe[i * Nn + j] = sr / d;
    xIm[i * Nn + j] = si / d;
  }
}

// ---------------------------------------------------------------------------
// Per-frequency Gauss-Jordan solve of [zI - M | RHS] + transfer function loss.
// ---------------------------------------------------------------------------
__global__ void __launch_bounds__(1024)
k_freq(const float* __restrict__ afRe, const float* __restrict__ afIm,
       const float* __restrict__ bfRe, const float* __restrict__ bfIm,
       const float* __restrict__ cfRe, const float* __restrict__ cfIm,
       const float* __restrict__ dfRe, const float* __restrict__ dfIm,
       const float* __restrict__ aRe,  const float* __restrict__ aIm,
       const float* __restrict__ cRe,  const float* __restrict__ cIm,
       const float* __restrict__ bhRe, const float* __restrict__ bhIm,
       float* __restrict__ slots, float* __restrict__ partial) {
  const int tid = threadIdx.x;
  const int nt = blockDim.x;
  float* augRe = slots + (size_t)blockIdx.x * SLOT_F;
  float* augIm = augRe + (size_t)Nn * NAUG;
  float* h1Re = augIm + (size_t)Nn * NAUG;
  float* h1Im = h1Re + Pp * Mm;

  __shared__ float prowRe[NAUG];
  __shared__ float prowIm[NAUG];
  __shared__ float pcolRe[Nn];
  __shared__ float pcolIm[Nn];
  __shared__ float red[1024];
  __shared__ float pivRe, pivIm;

  for (int f = blockIdx.x; f < NFREQ; f += gridDim.x) {
    float w = 6.283185307179586f * (float)f / (float)(NFREQ - 1);
    float zr = cosf(w), zi = sinf(w);
    float lossAcc = 0.0f;

    for (int pass = 0; pass < 2; ++pass) {
      const float* mRe = pass ? aRe : afRe;
      const float* mIm = pass ? aIm : afIm;
      const float* rRe = pass ? bhRe : bfRe;
      const float* rIm = pass ? bhIm : bfIm;

      for (int t = tid; t < Nn * NAUG; t += nt) {
        int i = t / NAUG, j = t % NAUG;
        float vr, vi;
        if (j < Nn) {
          vr = -mRe[i * Nn + j];
          vi = -mIm[i * Nn + j];
          if (i == j) { vr += zr; vi += zi; }
        } else {
          vr = rRe[i * Mm + (j - Nn)];
          vi = rIm[i * Mm + (j - Nn)];
        }
        augRe[t] = vr;
        augIm[t] = vi;
      }
      __syncthreads();

      for (int k = 0; k < Nn; ++k) {
        if (tid == 0) {
          float pr = augRe[k * NAUG + k], pi = augIm[k * NAUG + k];
          float s = 1.0f / (pr * pr + pi * pi);
          pivRe = pr * s;
          pivIm = -pi * s;
        }
        __syncthreads();
        for (int j = k + tid; j < NAUG; j += nt) {
          float vr = augRe[k * NAUG + j], vi = augIm[k * NAUG + j];
          float nr = vr * pivRe - vi * pivIm;
          float ni = vr * pivIm + vi * pivRe;
          augRe[k * NAUG + j] = nr;
          augIm[k * NAUG + j] = ni;
          prowRe[j] = nr;
          prowIm[j] = ni;
        }
        for (int i = tid; i < Nn; i += nt) {
          pcolRe[i] = augRe[i * NAUG + k];
          pcolIm[i] = augIm[i * NAUG + k];
        }
        __syncthreads();
        const int ty = tid >> 7;
        const int tx = tid & 127;
        for (int i = ty; i < Nn; i += 8) {
          if (i == k) continue;
          float fr = pcolRe[i], fi = pcolIm[i];
          for (int j = k + tx; j < NAUG; j += 128) {
            float pr = prowRe[j], pi = prowIm[j];
            augRe[i * NAUG + j] -= fr * pr - fi * pi;
            augIm[i * NAUG + j] -= fr * pi + fi * pr;
          }
        }
        __syncthreads();
      }

      const float* ccRe = pass ? cRe : cfRe;
      const float* ccIm = pass ? cIm : cfIm;
      for (int t = tid; t < Pp * Mm; t += nt) {
        int p = t / Mm, m = t % Mm;
        float sr = 0.0f, si = 0.0f;
        for (int n2 = 0; n2 < Nn; ++n2) {
          float ar = ccRe[p * Nn + n2], ai = ccIm[p * Nn + n2];
          float xr = augRe[n2 * NAUG + Nn + m], xi = augIm[n2 * NAUG + Nn + m];
          sr += ar * xr - ai * xi;
          si += ar * xi + ai * xr;
        }
        if (!pass) {
          sr += dfRe[t]; si += dfIm[t];
          h1Re[t] = sr; h1Im[t] = si;
        } else {
          float dr = h1Re[t] - sr, di = h1Im[t] - si;
          lossAcc += dr * dr + di * di;
        }
      }
      __syncthreads();
    }

    red[tid] = lossAcc;
    __syncthreads();
    for (int s = nt >> 1; s > 0; s >>= 1) {
      if (tid < s) red[tid] += red[tid + s];
      __syncthreads();
    }
    if (tid == 0) partial[f] = red[0];
    __syncthreads();
  }
}

__global__ void k_final(const float* __restrict__ partial, double* __restrict__ out) {
  if (threadIdx.x == 0 && blockIdx.x == 0) {
    float s = 0.0f;
    for (int f = 0; f < NFREQ; ++f) s += partial[f];
    out[0] = (double)(s / (float)NFREQ);
  }
}

// ---------------------------------------------------------------------------
// Host orchestration
// ---------------------------------------------------------------------------
extern "C" void kernel_launch(void* const* d_in, const int* in_sizes, int n_in,
                              void* d_out, int out_size, void* d_ws, size_t ws_size,
                              hipStream_t stream) {
  (void)in_sizes; (void)n_in; (void)out_size; (void)ws_size;
  const double* vParams = (const double*)d_in[0];
  const double* dAFre = (const double*)d_in[1];
  const double* dAFim = (const double*)d_in[2];
  const double* dBFre = (const double*)d_in[3];
  const double* dBFim = (const double*)d_in[4];
  const double* dCFre = (const double*)d_in[5];
  const double* dCFim = (const double*)d_in[6];
  const double* dDFre = (const double*)d_in[7];
  const double* dDFim = (const double*)d_in[8];

  float* w = (float*)d_ws;
  size_t off = 0;
  float* afRe = w + off; off += NN;  float* afIm = w + off; off += NN;
  float* bfRe = w + off; off += NM;  float* bfIm = w + off; off += NM;
  float* cfRe = w + off; off += PN;  float* cfIm = w + off; off += PN;
  float* dfRe = w + off; off += PM;  float* dfIm = w + off; off += PM;
  float* vRe  = w + off; off += PN;  float* vIm  = w + off; off += PN;
  float* gRe  = w + off; off += NN;  float* gIm  = w + off; off += NN;  // Gram -> R
  float* axRe = w + off; off += NN;  float* axIm = w + off; off += NN;  // A = R^{-1}
  float* cmRe = w + off; off += PN;  float* cmIm = w + off; off += PN;  // C = V R^{-1}
  float* rhRe = w + off; off += NN;  float* rhIm = w + off; off += NN;  // RHS
  float* qaRe = w + off; off += NN;  float* qaIm = w + off; off += NN;  // Q12 buf A
  float* qbRe = w + off; off += NN;  float* qbIm = w + off; off += NN;  // Q12 buf B
  float* tRe  = w + off; off += NN;  float* tIm  = w + off; off += NN;  // temp
  float* bhRe = w + off; off += NM;  float* bhIm = w + off; off += NM;  // B_hat
  float* slots = w + off; off += (size_t)NSLOT * SLOT_F;
  float* partial = w + off; off += 64;

  dim3 b256(256);
  k_d2f<<<(NN + 255) / 256, b256, 0, stream>>>(dAFre, afRe, NN);
  k_d2f<<<(NN + 255) / 256, b256, 0, stream>>>(dAFim, afIm, NN);
  k_d2f<<<(NM + 255) / 256, b256, 0, stream>>>(dBFre, bfRe, NM);
  k_d2f<<<(NM + 255) / 256, b256, 0, stream>>>(dBFim, bfIm, NM);
  k_d2f<<<(PN + 255) / 256, b256, 0, stream>>>(dCFre, cfRe, PN);
  k_d2f<<<(PN + 255) / 256, b256, 0, stream>>>(dCFim, cfIm, PN);
  k_d2f<<<(PM + 255) / 256, b256, 0, stream>>>(dDFre, dfRe, PM);
  k_d2f<<<(PM + 255) / 256, b256, 0, stream>>>(dDFim, dfIm, PM);
  k_buildV<<<(PN + 255) / 256, b256, 0, stream>>>(vParams, vRe, vIm, PN);

  // Cholesky-QR of [V; I]: G = I + V^H V ; R = chol(G) ; A = R^{-1} ; C = V R^{-1}
  k_cgemm<true><<<dim3(Nn / 32, Nn / 64), b256, 0, stream>>>(
      vRe, vIm, Nn, vRe, vIm, Nn, gRe, gIm, Nn, nullptr, nullptr, Pp, 1.0f);
  k_addeye<<<(Nn + 255) / 256, b256, 0, stream>>>(gRe);
  k_chol<<<1, 1024, 0, stream>>>(gRe, gIm);
  k_trinv<<<2, 256, 0, stream>>>(gRe, gIm, axRe, axIm);
  k_cgemm<false><<<dim3(Nn / 32, Pp / 64), b256, 0, stream>>>(
      vRe, vIm, Nn, axRe, axIm, Nn, cmRe, cmIm, Nn, nullptr, nullptr, Nn, 1.0f);

  // RHS = C_F^H @ C
  k_cgemm<true><<<dim3(Nn / 32, Nn / 64), b256, 0, stream>>>(
      cfRe, cfIm, Nn, cmRe, cmIm, Nn, rhRe, rhIm, Nn, nullptr, nullptr, Pp, 1.0f);

  // Stein iteration: Q12 <- A_F^H Q12 A + RHS  (50 iterations, 100 WMMA GEMMs)
  k_zero<<<(2 * NN + 255) / 256, b256, 0, stream>>>(qaRe, 2 * NN);
  float* curRe = qaRe; float* curIm = qaIm;
  float* nxtRe = qbRe; float* nxtIm = qbIm;
  for (int it = 0; it < NSTEIN; ++it) {
    k_cgemm<true><<<dim3(Nn / 32, Nn / 64), b256, 0, stream>>>(
        afRe, afIm, Nn, curRe, curIm, Nn, tRe, tIm, Nn, nullptr, nullptr, Nn, 1.0f);
    k_cgemm<false><<<dim3(Nn / 32, Nn / 64), b256, 0, stream>>>(
        tRe, tIm, Nn, axRe, axIm, Nn, nxtRe, nxtIm, Nn, rhRe, rhIm, Nn, 1.0f);
    float* sR = curRe; float* sI = curIm;
    curRe = nxtRe; curIm = nxtIm; nxtRe = sR; nxtIm = sI;
  }

  // B_hat = -(Q12^H @ B_F)
  k_cgemm<true><<<dim3(Mm / 32, Nn / 64), b256, 0, stream>>>(
      curRe, curIm, Nn, bfRe, bfIm, Mm, bhRe, bhIm, Mm, nullptr, nullptr, Nn, -1.0f);

  // Frequency-domain solves + loss
  k_freq<<<NSLOT, 1024, 0, stream>>>(afRe, afIm, bfRe, bfIm, cfRe, cfIm, dfRe, dfIm,
                                     axRe, axIm, cmRe, cmIm, bhRe, bhIm, slots, partial);
  k_final<<<1, 32, 0, stream>>>(partial, (double*)d_out);
}